// Gaussian_cross_attn_75342316306666
// MI455X (gfx1250) — compile-verified
//
#include <hip/hip_runtime.h>

// Gaussian cross-attn weights @ Q_emb  ->  out[64,512,2048] f32 (268 MB, store-BW bound)
// Matrix math on V_WMMA_F32_16X16X4_F32 (K=0..3), 5th weight folded into WMMA C operand.

typedef __attribute__((ext_vector_type(2))) float v2f;
typedef __attribute__((ext_vector_type(4))) float v4f;
typedef __attribute__((ext_vector_type(8))) float v8f;

#define B_DIM   64
#define M_DIM   512
#define D_DIM   2048
#define INV2VAR 0.08f   // 1/(2*2.5^2)

__global__ __launch_bounds__(256) void gauss_wmma_kernel(
    const int* __restrict__ seg,
    const float* __restrict__ Q,     // [B,5,D]
    float* __restrict__ out)         // [B,M,D]
{
    const int b    = blockIdx.y;
    const int m0   = blockIdx.x << 4;          // row-tile base (16 rows)
    const int tid  = threadIdx.x;
    const int lane = tid & 31;
    const int ln   = lane & 15;                // N index within tile / row index for weights
    const int hi   = lane >> 4;                // lane half
    const int wave = tid >> 5;

    const int nseg = seg[b];
    float* outB = out + ((size_t)b * M_DIM + (size_t)m0) * D_DIM;

    // ---- fast path: entire 16x2048 row-tile is past segment_len -> stream zeros ----
    if (m0 >= nseg) {
        v4f z = {0.f, 0.f, 0.f, 0.f};
        #pragma unroll 4
        for (int i = tid; i < 16 * (D_DIM / 4); i += 256) {
            __builtin_nontemporal_store(z, (v4f*)(outB + (size_t)i * 4));
        }
        return;
    }

    // ---- per-row Gaussian weights: row r = m0 + ln (both lane halves compute it) ----
    const int   r     = m0 + ln;
    const float nf    = (float)nseg;
    const float denom = fmaxf(nf - 1.0f, 1.0f);
    const float mu    = (float)r * (4.0f / denom);

    float w0, w1, w2, w3, w4;
    {
        float z0 = 0.0f - mu, z1 = 1.0f - mu, z2 = 2.0f - mu, z3 = 3.0f - mu, z4 = 4.0f - mu;
        float g0 = __expf(-z0 * z0 * INV2VAR);
        float g1 = __expf(-z1 * z1 * INV2VAR);
        float g2 = __expf(-z2 * z2 * INV2VAR);
        float g3 = __expf(-z3 * z3 * INV2VAR);
        float g4 = __expf(-z4 * z4 * INV2VAR);
        float gmin = fminf(fminf(fminf(g0, g1), fminf(g2, g3)), g4);
        float gmax = fmaxf(fmaxf(fmaxf(g0, g1), fmaxf(g2, g3)), g4);
        float s = 1.0f / (gmax - gmin);
        w0 = (g0 - gmin) * s; w1 = (g1 - gmin) * s; w2 = (g2 - gmin) * s;
        w3 = (g3 - gmin) * s; w4 = (g4 - gmin) * s;
        // reference `where` chain, in order:
        if (r == 0)               { w0 = 1.f; w1 = 0.f; w2 = 0.f; w3 = 0.f; w4 = 0.f; }
        if (r == nseg - 1)        { w0 = 0.f; w1 = 0.f; w2 = 0.f; w3 = 0.f; w4 = 1.f; }
        if (r >= nseg)            { w0 = 0.f; w1 = 0.f; w2 = 0.f; w3 = 0.f; w4 = 0.f; }
        if (nseg == 1 && r == 0)  { w0 = 1.f; w1 = 1.f; w2 = 1.f; w3 = 1.f; w4 = 1.f; }
    }

    // A operand, 16x4 f32 layout: VGPR0 = K0|K2, VGPR1 = K1|K3 across lane halves
    v2f a;
    a.x = hi ? w2 : w0;
    a.y = hi ? w3 : w1;

    // w4 broadcast for the C operand: element (j,lane) belongs to row m0 + j + 8*hi,
    // whose w4 lives in lane (j + 8*hi)  ->  8 bpermutes, hoisted out of the n-loop
    float w5j[8];
    #pragma unroll
    for (int j = 0; j < 8; ++j)
        w5j[j] = __shfl(w4, j + (hi << 3), 32);

    const float* Qb  = Q + (size_t)b * 5 * D_DIM;
    const float* q0p = Qb + (size_t)(2 * hi)     * D_DIM;  // B row K = 2*hi
    const float* q1p = Qb + (size_t)(2 * hi + 1) * D_DIM;  // B row K = 2*hi+1
    const float* q4p = Qb + (size_t)4 * D_DIM;
    float* outRow = outB + (size_t)(hi << 3) * D_DIM;      // D rows: j + 8*hi

    // 128 n-tiles of width 16, round-robined over 8 waves
    for (int nt = wave; nt < D_DIM / 16; nt += 8) {
        const int col = (nt << 4) + ln;

        v2f bv;
        bv.x = q0p[col];                 // L2-served reads (Q is 2.6MB total)
        bv.y = q1p[col];
        const float q4 = q4p[col];

        v8f c;
        #pragma unroll
        for (int j = 0; j < 8; ++j) c[j] = w5j[j] * q4;   // fold 5th k into accumulator

        v8f d = __builtin_amdgcn_wmma_f32_16x16x4_f32(
            /*neg_a=*/false, a, /*neg_b=*/false, bv,
            /*c_mod=*/(short)0, c, /*reuse_a=*/false, /*reuse_b=*/false);

        #pragma unroll
        for (int j = 0; j < 8; ++j)
            __builtin_nontemporal_store(d[j], outRow + (size_t)j * D_DIM + col);
    }
}

extern "C" void kernel_launch(void* const* d_in, const int* in_sizes, int n_in,
                              void* d_out, int out_size, void* d_ws, size_t ws_size,
                              hipStream_t stream) {
    const int*   seg = (const int*)d_in[0];     // segment_len [64] int32
    // d_in[1] = max_chunk scalar (512) -- fixed by the reference shapes
    const float* Q   = (const float*)d_in[2];   // Q_emb [64,5,2048] f32
    float*       out = (float*)d_out;           // [64,512,2048] f32

    dim3 grid(M_DIM / 16, B_DIM);               // (32 row-tiles, 64 batches)
    gauss_wmma_kernel<<<grid, 256, 0, stream>>>(seg, Q, out);
}